// BusinessCostLoss_52819507806485
// MI455X (gfx1250) — compile-verified
//
#include <hip/hip_runtime.h>

// ---------------------------------------------------------------------------
// BusinessCostLoss on MI455X (gfx1250, wave32)
//
// Streaming cost-weighted CE loss over 3 tasks of [8M,2] logits + [8M] labels.
// Memory-bound: 288 MiB read-once => ~12.4us floor @ 23.3 TB/s.
//  - b128 non-temporal loads (stream > 192MB L2, zero reuse)
//  - hardware transcendentals (v_exp_f32 / v_log_f32)
//  - wave reduction via V_WMMA_F32_16X16X4_F32 (A = lane partials, B = ones)
//  - deterministic two-stage reduction (no float atomics)
// ---------------------------------------------------------------------------

typedef float v2f  __attribute__((ext_vector_type(2)));
typedef float v4f  __attribute__((ext_vector_type(4)));
typedef float v8f  __attribute__((ext_vector_type(8)));
typedef int   v4i  __attribute__((ext_vector_type(4)));

#define NBLOCKS 2048
#define NTHREADS 256

// Per-element loss:
//   p_true = softmax(l)[y]  (stable sigmoid of logit diff)
//   base   = -log(p_true + 1e-8)
//   pred   = argmax(l)  (first index on tie -> pred = (l1 > l0))
//   w      = pred==y ? 0.1 : (y==1 ? 5.0 : 1.0)   // miss=5, false alarm=1
__device__ __forceinline__ float elem_loss(float l0, float l1, int y) {
    float d   = (y != 0) ? (l1 - l0) : (l0 - l1);     // l_true - l_other
    float t   = __expf(-fabsf(d));                    // v_exp_f32
    float inv = 1.0f / (1.0f + t);
    float p   = (d >= 0.0f) ? inv : t * inv;          // stable sigmoid
    float base = -__logf(p + 1e-8f);                  // v_log_f32
    int  pred  = (l1 > l0) ? 1 : 0;
    float w    = (pred == y) ? 0.1f : ((y != 0) ? 5.0f : 1.0f);
    return w * base;
}

// Sum one f32 partial across all 32 lanes of the wave using
// V_WMMA_F32_16X16X4_F32:  A = [p_m | 0 | p_{m+16} | 0] (16x4), B = ones (4x16)
//   => D[j][n] = p_j + p_{j+16}.  C/D layout: lane n<16 holds rows 0..7 in its
// 8 D-VGPRs, lane n>=16 holds rows 8..15.  So sum(D VGPRs) + shfl_xor(16)
// yields the full 32-lane sum in every lane, entirely in f32.
__device__ __forceinline__ float wave_reduce_wmma(float p) {
    v2f a; a.x = p;    a.y = 0.0f;
    v2f b; b.x = 1.0f; b.y = 1.0f;
    v8f c = {};
    v8f d = __builtin_amdgcn_wmma_f32_16x16x4_f32(
        /*neg_a=*/false, a, /*neg_b=*/false, b,
        /*c_mod=*/(short)0, c, /*reuse_a=*/false, /*reuse_b=*/false);
    float s = ((d[0] + d[1]) + (d[2] + d[3])) + ((d[4] + d[5]) + (d[6] + d[7]));
    s += __shfl_xor(s, 16, 32);
    return s;
}

__global__ __launch_bounds__(NTHREADS)
void stage1_partials(const float* __restrict__ La, const float* __restrict__ Lb,
                     const float* __restrict__ Lc,
                     const int*  __restrict__ Ta, const int* __restrict__ Tb,
                     const int*  __restrict__ Tc,
                     float* __restrict__ partials, int n4, int nblocks) {
    float accA = 0.0f, accB = 0.0f, accC = 0.0f;
    const int stride = gridDim.x * blockDim.x;

    for (int i = blockIdx.x * blockDim.x + threadIdx.x; i < n4; i += stride) {
        const int e = i << 2;          // first element of this 4-element chunk
        // ---- task A ----
        {
            v4i y  = __builtin_nontemporal_load((const v4i*)(Ta + e));
            v4f x0 = __builtin_nontemporal_load((const v4f*)(La + 2 * e));
            v4f x1 = __builtin_nontemporal_load((const v4f*)(La + 2 * e + 4));
            accA += elem_loss(x0.x, x0.y, y.x) + elem_loss(x0.z, x0.w, y.y)
                  + elem_loss(x1.x, x1.y, y.z) + elem_loss(x1.z, x1.w, y.w);
        }
        // ---- task B ----
        {
            v4i y  = __builtin_nontemporal_load((const v4i*)(Tb + e));
            v4f x0 = __builtin_nontemporal_load((const v4f*)(Lb + 2 * e));
            v4f x1 = __builtin_nontemporal_load((const v4f*)(Lb + 2 * e + 4));
            accB += elem_loss(x0.x, x0.y, y.x) + elem_loss(x0.z, x0.w, y.y)
                  + elem_loss(x1.x, x1.y, y.z) + elem_loss(x1.z, x1.w, y.w);
        }
        // ---- task C ----
        {
            v4i y  = __builtin_nontemporal_load((const v4i*)(Tc + e));
            v4f x0 = __builtin_nontemporal_load((const v4f*)(Lc + 2 * e));
            v4f x1 = __builtin_nontemporal_load((const v4f*)(Lc + 2 * e + 4));
            accC += elem_loss(x0.x, x0.y, y.x) + elem_loss(x0.z, x0.w, y.y)
                  + elem_loss(x1.x, x1.y, y.z) + elem_loss(x1.z, x1.w, y.w);
        }
    }

    // Wave-level reduction via WMMA (all 32 lanes active; EXEC == all 1s).
    float wa = wave_reduce_wmma(accA);
    float wb = wave_reduce_wmma(accB);
    float wc = wave_reduce_wmma(accC);

    // Cross-wave reduction in LDS (8 wave32s per block).
    __shared__ float wsum[3][NTHREADS / 32];
    const int lane = threadIdx.x & 31;
    const int wid  = threadIdx.x >> 5;
    if (lane == 0) {
        wsum[0][wid] = wa;
        wsum[1][wid] = wb;
        wsum[2][wid] = wc;
    }
    __syncthreads();
    if (threadIdx.x < 3) {
        float s = 0.0f;
#pragma unroll
        for (int w = 0; w < NTHREADS / 32; ++w) s += wsum[threadIdx.x][w];
        partials[threadIdx.x * nblocks + blockIdx.x] = s;
    }
}

__global__ __launch_bounds__(NTHREADS)
void stage2_finalize(const float* __restrict__ partials, int nblocks, int n,
                     float* __restrict__ out) {
    __shared__ double red[NTHREADS];
    __shared__ double mean[3];
    const double invN = 1.0 / (double)n;

    for (int t = 0; t < 3; ++t) {
        double s = 0.0;
        for (int i = threadIdx.x; i < nblocks; i += NTHREADS)
            s += (double)partials[t * nblocks + i];
        red[threadIdx.x] = s;
        __syncthreads();
#pragma unroll
        for (int off = NTHREADS / 2; off > 0; off >>= 1) {
            if ((int)threadIdx.x < off) red[threadIdx.x] += red[threadIdx.x + off];
            __syncthreads();
        }
        if (threadIdx.x == 0) mean[t] = red[0] * invN;
        __syncthreads();
    }

    if (threadIdx.x == 0) {
        float la = (float)mean[0];
        float lb = (float)mean[1];
        float lc = (float)mean[2];
        out[0] = la;
        out[1] = lb;
        out[2] = lc;
        out[3] = 1.0f * la + 0.5f * lb + 2.0f * lc;   // TASK_WEIGHTS
    }
}

extern "C" void kernel_launch(void* const* d_in, const int* in_sizes, int n_in,
                              void* d_out, int out_size, void* d_ws, size_t ws_size,
                              hipStream_t stream) {
    const float* La = (const float*)d_in[0];
    const float* Lb = (const float*)d_in[1];
    const float* Lc = (const float*)d_in[2];
    const int*   Ta = (const int*)d_in[3];
    const int*   Tb = (const int*)d_in[4];
    const int*   Tc = (const int*)d_in[5];
    float* out      = (float*)d_out;
    float* partials = (float*)d_ws;            // 3 * NBLOCKS floats = 24 KiB

    const int n  = in_sizes[3];                // B = 8388608 (per task)
    const int n4 = n >> 2;                     // 4 elements per chunk

    stage1_partials<<<NBLOCKS, NTHREADS, 0, stream>>>(La, Lb, Lc, Ta, Tb, Tc,
                                                      partials, n4, NBLOCKS);
    stage2_finalize<<<1, NTHREADS, 0, stream>>>(partials, NBLOCKS, n, out);
}